// EdgeLLMAttentionTRTNative_29360396436022
// MI455X (gfx1250) — compile-verified
//
#include <hip/hip_runtime.h>

// ---------------------------------------------------------------------------
// Reference math (dead code removed):
//   out  = (128^-0.5) * (X @ Wq^T) @ Wo^T      X: (8192, 3072) fp32
//        = (128^-0.5) *  X @ (Wo @ Wq)^T       (weight folding: 31% fewer FLOPs)
//   kv_cache passes through unchanged (128 MB copy).
// Compute-bound fp32 GEMM -> use V_WMMA_F32_16X16X4_F32 (exact fp32 on the
// CDNA5 matrix pipe).
// ---------------------------------------------------------------------------

typedef __attribute__((ext_vector_type(2))) float v2f;
typedef __attribute__((ext_vector_type(4))) float v4f;
typedef __attribute__((ext_vector_type(8))) float v8f;

#define KC      16   // K-chunk staged in LDS (4 WMMA k-steps of 4)
#define LDA_PAD 20   // A-tile row stride (floats): 80B = 16B-aligned, bank-safe
#define LDBT    20   // B-tile row stride for [N][K] layout
#define LDBK    72   // B-tile row stride for [K][N] layout (288B, 16B-aligned)

// C[m,n] = alpha * sum_k A[m,k] * Bop(k,n)
//   BT=true : B stored [N][K] row-major (nn.Linear weight), Bop(k,n)=B[n*K+k]
//   BT=false: B stored [K][N] row-major,                    Bop(k,n)=B[k*N+n]
// Tile: workgroup = 256 threads = 8 waves -> 128(M) x 64(N); wave -> 32x32.
template <bool BT>
__global__ __launch_bounds__(256) void gemm_wmma_f32(
    const float* __restrict__ A, const float* __restrict__ B,
    float* __restrict__ C, int M, int N, int K, float alpha)
{
    __shared__ __attribute__((aligned(16))) float As[128 * LDA_PAD];
    __shared__ __attribute__((aligned(16))) float Bs[BT ? (64 * LDBT) : (KC * LDBK)];

    const int tid  = threadIdx.x;
    const int lane = tid & 31;
    const int wave = tid >> 5;
    const int wm   = (wave & 3) * 32;   // wave's M offset inside the 128-row tile
    const int wn   = (wave >> 2) * 32;  // wave's N offset inside the 64-col tile
    const int m0   = blockIdx.x * 128;
    const int n0   = blockIdx.y * 64;
    const int lr   = lane & 15;         // fragment row (A) / col (B,C)
    const int lh   = lane >> 4;         // half-wave selector

    v8f acc[2][2] = {};                 // 4x 16x16 f32 accumulators (32 VGPRs)

    for (int k0 = 0; k0 < K; k0 += KC) {
        // ---- stage A tile: 128 rows x 16 floats (512 x b128, 2 per thread) ----
        #pragma unroll
        for (int r = 0; r < 2; ++r) {
            int idx = tid + r * 256;
            int row = idx >> 2;
            int c4  = (idx & 3) << 2;
            v4f a = *(const v4f*)(A + (size_t)(m0 + row) * K + k0 + c4);
            *(v4f*)&As[row * LDA_PAD + c4] = a;
        }
        // ---- stage B tile (256 x b128, 1 per thread) ----
        if (BT) {
            int row = tid >> 2;
            int c4  = (tid & 3) << 2;
            v4f b = *(const v4f*)(B + (size_t)(n0 + row) * K + k0 + c4);
            *(v4f*)&Bs[row * LDBT + c4] = b;
        } else {
            int row = tid >> 4;
            int c4  = (tid & 15) << 2;
            v4f b = *(const v4f*)(B + (size_t)(k0 + row) * N + n0 + c4);
            *(v4f*)&Bs[row * LDBK + c4] = b;
        }
        // ---- prefetch next K-chunk while this one computes ----
        if (k0 + KC < K) {
            __builtin_prefetch(A + (size_t)(m0 + (tid & 127)) * K + (k0 + KC) + (tid >> 7) * 8, 0, 1);
            if (BT) {
                if (tid < 64) __builtin_prefetch(B + (size_t)(n0 + tid) * K + (k0 + KC), 0, 1);
            } else {
                if (tid < 16) __builtin_prefetch(B + (size_t)(k0 + KC + tid) * N + n0, 0, 1);
            }
        }
        __syncthreads();

        // ---- 4 WMMA k-steps of 4 over the chunk ----
        #pragma unroll
        for (int ks = 0; ks < KC; ks += 4) {
            v2f a2[2], b2[2];
            #pragma unroll
            for (int mi = 0; mi < 2; ++mi)
                a2[mi] = *(const v2f*)&As[(wm + mi * 16 + lr) * LDA_PAD + ks + lh * 2];
            #pragma unroll
            for (int ni = 0; ni < 2; ++ni) {
                if (BT) {
                    b2[ni] = *(const v2f*)&Bs[(wn + ni * 16 + lr) * LDBT + ks + lh * 2];
                } else {
                    int rr = ks + lh * 2;
                    v2f b;
                    b.x = Bs[(rr + 0) * LDBK + wn + ni * 16 + lr];
                    b.y = Bs[(rr + 1) * LDBK + wn + ni * 16 + lr];
                    b2[ni] = b;
                }
            }
            #pragma unroll
            for (int mi = 0; mi < 2; ++mi)
                #pragma unroll
                for (int ni = 0; ni < 2; ++ni)
                    acc[mi][ni] = __builtin_amdgcn_wmma_f32_16x16x4_f32(
                        false, a2[mi], false, b2[ni],
                        (short)0, acc[mi][ni], false, false);
        }
        __syncthreads();
    }

    // ---- epilogue: C/D layout — VGPR r, lane L -> (row r + 8*(L>=16), col L%16) ----
    #pragma unroll
    for (int mi = 0; mi < 2; ++mi)
        #pragma unroll
        for (int ni = 0; ni < 2; ++ni) {
            int col   = n0 + wn + ni * 16 + lr;
            int rbase = m0 + wm + mi * 16 + lh * 8;
            float* cp = C + (size_t)rbase * N + col;
            #pragma unroll
            for (int r = 0; r < 8; ++r)
                cp[(size_t)r * N] = alpha * acc[mi][ni][r];
        }
}

// 128 MB kv_cache passthrough: grid-stride float4 copy (bandwidth-bound).
__global__ __launch_bounds__(256) void copy_f4(const float4* __restrict__ src,
                                               float4* __restrict__ dst, long n4)
{
    long i = (long)blockIdx.x * 256 + threadIdx.x;
    long stride = (long)gridDim.x * 256;
    for (; i < n4; i += stride) dst[i] = src[i];
}

extern "C" void kernel_launch(void* const* d_in, const int* in_sizes, int n_in,
                              void* d_out, int out_size, void* d_ws, size_t ws_size,
                              hipStream_t stream)
{
    (void)in_sizes; (void)n_in; (void)out_size;

    const float* X  = (const float*)d_in[0];  // hidden_states (4,2048,3072)
    const float* kv = (const float*)d_in[4];  // kv_cache (4,2,8,4096,128)
    const float* Wq = (const float*)d_in[5];  // (3072, 3072)
    const float* Wo = (const float*)d_in[8];  // (3072, 3072)
    // d_in[1..3], d_in[6..7] (rope, lengths, Wk, Wv) are dead w.r.t. outputs.

    float* out = (float*)d_out;
    const long M = 8192, N = 3072, K = 3072;
    const long out_elems = M * N;                       // 25,165,824
    const long kv_elems  = 4L * 2 * 8 * 4096 * 128;     // 33,554,432
    float* kv_out = out + out_elems;

    // Weff scratch (36 MB): d_ws if big enough, else transiently in the kv
    // region of d_out (overwritten by the kv copy, which runs last in-order).
    size_t weff_bytes = (size_t)N * K * sizeof(float);
    float* weff = (ws_size >= weff_bytes) ? (float*)d_ws : kv_out;

    dim3 blk(256);
    // Weff = Wo @ Wq : A=Wo (rows o, k=d), B=Wq ([d][h] = [K][N] layout)
    gemm_wmma_f32<false><<<dim3(3072 / 128, 3072 / 64), blk, 0, stream>>>(
        Wo, Wq, weff, 3072, 3072, 3072, 1.0f);
    // out = scale * X @ Weff^T : Weff in [N][K] (nn.Linear) layout
    const float scale = 0.08838834764831845f;  // 128^-0.5
    gemm_wmma_f32<true><<<dim3(8192 / 128, 3072 / 64), blk, 0, stream>>>(
        X, weff, out, 8192, 3072, 3072, scale);
    // kv passthrough
    copy_f4<<<dim3(8192), blk, 0, stream>>>((const float4*)kv, (float4*)kv_out, kv_elems / 4);
}